// SuperLoss_32804960206851
// MI455X (gfx1250) — compile-verified
//
#include <hip/hip_runtime.h>

// ---------------------------------------------------------------------------
// SuperLoss forward (mean) for MI455X / gfx1250.
//   Stage 1: grid-stride stream of fp32 losses (GLOBAL_LOAD_B128 +
//            global_prefetch_b8), per-element Lambert-W via branchless Halley
//            (native v_exp/v_log/v_rcp; cubic convergence makes fast-math init
//            and rcp-division exact at the fixed point), summation routed
//            through V_WMMA_F32_16X16X4_F32 with a ones-B matrix
//            (64 values folded per WMMA; row sums replicated x16 in D).
//   Stage 2: deterministic reduction of per-block partials -> mean.
// ---------------------------------------------------------------------------

typedef __attribute__((ext_vector_type(2))) float v2f;
typedef __attribute__((ext_vector_type(8))) float v8f;

#define NBLK 2048
#define NTHR 256
#define HALLEY_ITERS 8

__device__ __forceinline__ float fast_rcp(float x) {
    return __builtin_amdgcn_rcpf(x);  // v_rcp_f32
}

__device__ __forceinline__ float superloss_elem(float l) {
    const float tau     = 6.90775527898f;   // log(1000)
    const float inv_lam = 4.0f;             // 1/0.25
    const float lam     = 0.25f;
    const float xmin    = -0.73575888234f;  // -2/e
    const float e1      = 2.71828182846f;

    float y = 0.5f * fmaxf(xmin, (l - tau) * inv_lam);   // y >= -1/e

    // Branchless init: branch-point expansion (y<0) vs fast log (y>=0).
    // Init error is irrelevant: Halley converges cubically to the true root.
    float p     = __builtin_amdgcn_sqrtf(fmaxf(2.0f * (1.0f + e1 * y), 0.0f));
    float w_neg = -1.0f + p - p * p * 0.33333333333f;
    float w_pos = __logf(1.0f + y);          // v_log_f32 path, no slow fixups
    float w     = (y < 0.0f) ? w_neg : w_pos;

    // Halley, one v_rcp per iteration:
    //   delta = 2*safe*f * rcp(2*safe^2*ew - (w+2)*f)
#pragma unroll
    for (int i = 0; i < HALLEY_ITERS; ++i) {
        float ew   = __expf(w);              // v_exp_f32
        float f    = w * ew - y;
        float wp1  = w + 1.0f;
        float safe = (fabsf(wp1) < 1e-6f) ? 1e-6f : wp1;
        float den  = 2.0f * safe * safe * ew - (w + 2.0f) * f;
        w          = w - (2.0f * safe * f) * fast_rcp(den);
    }

    float sigma = __expf(-w);
    // log(sigma) = -w  =>  lam*log(sigma)^2 = lam*w*w
    return (l - tau) * sigma + lam * w * w;
}

__global__ void __launch_bounds__(NTHR)
superloss_stage1(const float* __restrict__ l, float* __restrict__ partial, int n) {
    const long long tid    = (long long)blockIdx.x * NTHR + threadIdx.x;
    const long long total4 = (long long)NBLK * NTHR * 4;
    const int       iters  = (int)(n / total4);  // uniform -> EXEC all-1s for WMMA

    v8f       c    = {};
    const v2f ones = {1.0f, 1.0f};

    long long base = tid * 4;
    for (int it = 0; it < iters; ++it) {
        const float4 v = *(const float4*)(l + base);
        __builtin_prefetch(l + base + total4, 0, 1);  // global_prefetch_b8

        float s0 = superloss_elem(v.x);
        float s1 = superloss_elem(v.y);
        float s2 = superloss_elem(v.z);
        float s3 = superloss_elem(v.w);

        v2f a0 = {s0, s1};
        v2f a1 = {s2, s3};
        // D = A x ones + C : every D row holds Sum_k A[row,k], replicated in
        // 16 columns -> final total is 16x inflated, undone in stage 2.
        c = __builtin_amdgcn_wmma_f32_16x16x4_f32(false, a0, false, ones,
                                                  (short)0, c, false, false);
        c = __builtin_amdgcn_wmma_f32_16x16x4_f32(false, a1, false, ones,
                                                  (short)0, c, false, false);
        base += total4;
    }

    float lane = c[0] + c[1] + c[2] + c[3] + c[4] + c[5] + c[6] + c[7];

    // Scalar tail (empty for N=2^25); x16 to match WMMA-inflated sums.
    for (long long i = (long long)iters * total4 + tid; i < n;
         i += (long long)NBLK * NTHR)
        lane += 16.0f * superloss_elem(l[i]);

    // wave32 butterfly reduction
    for (int off = 16; off > 0; off >>= 1)
        lane += __shfl_xor(lane, off, 32);

    __shared__ float smem[NTHR / 32];
    const int lane_id = threadIdx.x & 31;
    const int wave_id = threadIdx.x >> 5;
    if (lane_id == 0) smem[wave_id] = lane;
    __syncthreads();

    if (threadIdx.x == 0) {
        float s = 0.0f;
#pragma unroll
        for (int i = 0; i < NTHR / 32; ++i) s += smem[i];
        partial[blockIdx.x] = s;
    }
}

__global__ void __launch_bounds__(256)
superloss_stage2(const float* __restrict__ partial, float* __restrict__ out,
                 int nblk, int n) {
    float s = 0.0f;
    for (int i = threadIdx.x; i < nblk; i += 256) s += partial[i];

    for (int off = 16; off > 0; off >>= 1)
        s += __shfl_xor(s, off, 32);

    __shared__ float smem[8];
    const int lane_id = threadIdx.x & 31;
    const int wave_id = threadIdx.x >> 5;
    if (lane_id == 0) smem[wave_id] = s;
    __syncthreads();

    if (threadIdx.x == 0) {
        float t = 0.0f;
#pragma unroll
        for (int i = 0; i < 8; ++i) t += smem[i];
        out[0] = t * (1.0f / 16.0f) / (float)n;  // undo x16 WMMA replication, mean
    }
}

extern "C" void kernel_launch(void* const* d_in, const int* in_sizes, int n_in,
                              void* d_out, int out_size, void* d_ws, size_t ws_size,
                              hipStream_t stream) {
    const float* l   = (const float*)d_in[0];
    float*       out = (float*)d_out;
    float*       ws  = (float*)d_ws;  // NBLK fp32 partials (8 KB)
    const int    n   = in_sizes[0];

    superloss_stage1<<<NBLK, NTHR, 0, stream>>>(l, ws, n);
    superloss_stage2<<<1, 256, 0, stream>>>(ws, out, NBLK, n);
}